// ContentAddressing_83159156785501
// MI455X (gfx1250) — compile-verified
//
#include <hip/hip_runtime.h>
#include <math.h>

typedef __attribute__((ext_vector_type(2))) float v2f;
typedef __attribute__((ext_vector_type(8))) float v8f;

#define B_   8
#define H_   8
#define C_   8192
#define W_   256
#define EPSF 1e-8f

// ---------------------------------------------------------------------------
// Kernel 1: scores[b,h,c] = dots/(||k_m||*||m_m|| + eps) * beta  -> d_out
// One block per (batch, 128-row tile): 8 waves x 16 rows.
// Uses V_WMMA_F32_16X16X4_F32: A = memory rows (16x4), B = km2/m2 (4x16,
// heads padded to 16 with zeros), accumulate over W in steps of K=4.
// ---------------------------------------------------------------------------
__global__ __launch_bounds__(256) void ca_scores_kernel(
    const float* __restrict__ memory,   // [B, C, W]
    const float* __restrict__ keys,     // [B, H, W]
    const float* __restrict__ betas,    // [B, H]
    const float* __restrict__ mask,     // [B, H, W]
    float* __restrict__ out)            // [B, H, C] (scores, pre-softmax)
{
    __shared__ float lds_km2[16][W_];   // keys*mask^2, rows 8..15 = 0
    __shared__ float lds_m2 [16][W_];   // mask^2,      rows 8..15 = 0
    __shared__ float red    [H_][W_];   // (keys*mask)^2 for key-norm reduce
    __shared__ float lds_knorm[H_];
    __shared__ float lds_beta [H_];

    const int b    = blockIdx.x >> 6;   // 64 tiles per batch
    const int tile = blockIdx.x & 63;
    const int tid  = threadIdx.x;

    // ---- prologue: build B-operands in LDS --------------------------------
    {
        const int w = tid;              // 0..255
        #pragma unroll
        for (int h = 0; h < H_; ++h) {
            float k   = keys[((size_t)(b * H_ + h)) * W_ + w];
            float m   = mask[((size_t)(b * H_ + h)) * W_ + w];
            float m2v = m * m;
            lds_km2[h][w]      = k * m2v;
            lds_m2 [h][w]      = m2v;
            lds_km2[h + 8][w]  = 0.0f;  // zero-pad heads 8..15 (N=16 tile)
            lds_m2 [h + 8][w]  = 0.0f;
            float km = k * m;
            red[h][w] = km * km;
        }
    }
    __syncthreads();
    if (tid < H_) {
        float s = 0.0f;
        for (int w = 0; w < W_; ++w) s += red[tid][w];
        lds_knorm[tid] = sqrtf(s);
        lds_beta [tid] = betas[b * H_ + tid];
    }
    __syncthreads();

    // ---- main WMMA loop ---------------------------------------------------
    const int wave   = tid >> 5;
    const int lane   = tid & 31;
    const int mrow   = lane & 15;              // A: M row / B: N col
    const int khalf  = (lane >> 4) << 1;       // K offset within 4: 0 or 2
    const int c_base = tile * 128 + wave * 16;

    const float* aptr = memory + ((size_t)b * C_ + (c_base + mrow)) * W_ + khalf;
    const float* bkp  = &lds_km2[mrow][khalf];
    const float* bmp  = &lds_m2 [mrow][khalf];

    v8f dacc = {};   // dots accumulator (16x16 f32)
    v8f sacc = {};   // squared-dots accumulator

    #pragma unroll 4
    for (int w = 0; w < W_; w += 4) {
        v2f a   = *(const v2f*)(aptr + w);     // A frag: K=khalf, khalf+1
        v2f a2  = a * a;                       // A^2 frag
        v2f bkm = *(const v2f*)(bkp + w);      // B frag (km2)
        v2f bm2 = *(const v2f*)(bmp + w);      // B frag (m2)
        // D = A x B + C   (8 args: neg_a, A, neg_b, B, c_mod, C, reuse_a, reuse_b)
        dacc = __builtin_amdgcn_wmma_f32_16x16x4_f32(
            false, a,  false, bkm, (short)0, dacc, false, false);
        sacc = __builtin_amdgcn_wmma_f32_16x16x4_f32(
            false, a2, false, bm2, (short)0, sacc, false, false);
    }

    // ---- epilogue: score = dots / (knorm * sqrt(sq) + eps) * beta ---------
    const int head  = lane & 15;                       // N = head
    const int rbase = c_base + ((lane >> 4) << 3);     // M = i + 8*(lane/16)
    if (head < H_) {
        const float kn = lds_knorm[head];
        const float bt = lds_beta [head];
        float* orow = out + ((size_t)(b * H_ + head)) * C_;
        #pragma unroll
        for (int i = 0; i < 8; ++i) {
            float denom = kn * sqrtf(sacc[i]) + EPSF;
            orow[rbase + i] = dacc[i] / denom * bt;
        }
    }
}

// ---------------------------------------------------------------------------
// Kernel 2: in-place softmax over C for each of the B*H = 64 rows.
// Row (8192 f32) held in registers: load once, max-reduce, exp+sum, scale.
// ---------------------------------------------------------------------------
__global__ __launch_bounds__(256) void ca_softmax_kernel(float* __restrict__ out)
{
    __shared__ float sred[256];
    const int row = blockIdx.x;           // 0..63
    const int tid = threadIdx.x;
    float* p = out + (size_t)row * C_;

    float v[32];
    float mx = -INFINITY;
    #pragma unroll
    for (int j = 0; j < 32; ++j) {
        v[j] = p[tid + j * 256];
        mx = fmaxf(mx, v[j]);
    }
    sred[tid] = mx;
    __syncthreads();
    #pragma unroll
    for (int s = 128; s > 0; s >>= 1) {
        if (tid < s) sred[tid] = fmaxf(sred[tid], sred[tid + s]);
        __syncthreads();
    }
    mx = sred[0];
    __syncthreads();

    float sum = 0.0f;
    #pragma unroll
    for (int j = 0; j < 32; ++j) {
        v[j] = __expf(v[j] - mx);
        sum += v[j];
    }
    sred[tid] = sum;
    __syncthreads();
    #pragma unroll
    for (int s = 128; s > 0; s >>= 1) {
        if (tid < s) sred[tid] += sred[tid + s];
        __syncthreads();
    }
    const float inv = 1.0f / sred[0];
    #pragma unroll
    for (int j = 0; j < 32; ++j) p[tid + j * 256] = v[j] * inv;
}

// ---------------------------------------------------------------------------
extern "C" void kernel_launch(void* const* d_in, const int* in_sizes, int n_in,
                              void* d_out, int out_size, void* d_ws, size_t ws_size,
                              hipStream_t stream) {
    (void)in_sizes; (void)n_in; (void)out_size; (void)d_ws; (void)ws_size;
    const float* memory = (const float*)d_in[0];
    const float* keys   = (const float*)d_in[1];
    const float* betas  = (const float*)d_in[2];
    const float* mask   = (const float*)d_in[3];
    float* out = (float*)d_out;

    ca_scores_kernel<<<dim3(B_ * 64), dim3(256), 0, stream>>>(
        memory, keys, betas, mask, out);
    ca_softmax_kernel<<<dim3(B_ * H_), dim3(256), 0, stream>>>(out);
}